// LFABlock_65532611002531
// MI455X (gfx1250) — compile-verified
//
#include <hip/hip_runtime.h>

typedef __attribute__((ext_vector_type(2))) float v2f;
typedef __attribute__((ext_vector_type(8))) float v8f;

#define LEAKY_SLOPE 0.2f

constexpr int NB   = 4;      // batches
constexpr int NPTS = 16384;  // points per batch
constexpr int FD   = 64;     // feature dim
constexpr int KNN  = 16;     // neighbors

// One wave32 per point.
//  - lanes gather neighbor geo (rel xyz + dist) into WMMA A layout (16x4 f32)
//  - 4x v_wmma_f32_16x16x4_f32 against preloaded W column tiles (4x16)
//  - bias + leaky-relu elementwise on D, reduce over M (=K neighbors) via
//    8-VGPR sum + shfl_xor(16), scale by 1/16
//  - feature mean: neighbor indices broadcast to SGPRs via readlane so the
//    16 coalesced 256B row gathers use scalar-base (saddr) addressing
__global__ __launch_bounds__(256) void lfa_kernel(
    const float* __restrict__ points,    // [NB, NPTS, 3]
    const float* __restrict__ features,  // [NB, NPTS, 64]
    const int*  __restrict__ knn,        // [NB, NPTS, 16]
    const float* __restrict__ W,         // [4, 64]
    const float* __restrict__ bias,      // [64]
    float* __restrict__ out)             // [NB, NPTS, 128]
{
    const int lane = threadIdx.x & 31;
    const int wave = threadIdx.x >> 5;
    const int p    = blockIdx.x * 8 + wave;   // global point id, [0, NB*NPTS)
    const int b    = p >> 14;                 // p / NPTS
    const int nn   = lane & 15;
    const int khi  = lane >> 4;               // 0: K-dims {0,1}, 1: K-dims {2,3}

    // ---- W column tiles in WMMA B layout (B is 4x16 per tile, 2 VGPRs) ----
    // lane (khi, nn): vgpr0 = W[2*khi][t*16+nn], vgpr1 = W[2*khi+1][t*16+nn]
    v2f  wt[4];
    float bv[4];
#pragma unroll
    for (int t = 0; t < 4; ++t) {
        wt[t].x = W[(2 * khi)     * FD + t * 16 + nn];
        wt[t].y = W[(2 * khi + 1) * FD + t * 16 + nn];
        bv[t]   = bias[t * 16 + nn];
    }

    // ---- gather geo for neighbor m = lane & 15 (both wave halves load it) ----
    const int  idx   = knn[(long)p * KNN + nn];
    const long pbase = (long)p * 3;
    const float cx = points[pbase + 0];
    const float cy = points[pbase + 1];
    const float cz = points[pbase + 2];
    const long nbase = ((long)b * NPTS + idx) * 3;
    const float rx = cx - points[nbase + 0];
    const float ry = cy - points[nbase + 1];
    const float rz = cz - points[nbase + 2];
    const float dist = __builtin_amdgcn_sqrtf(rx * rx + ry * ry + rz * rz);

    // WMMA A layout (16x4 f32): lanes 0-15 hold K={0,1}, lanes 16-31 K={2,3}
    v2f a;
    a.x = khi ? rz   : rx;
    a.y = khi ? dist : ry;

    // ---- feature mean: indices to SGPRs, scalar-base coalesced row gathers ----
    // Each lane accumulates channels (2*lane, 2*lane+1); 256B per neighbor row.
    const float2* __restrict__ fbase = (const float2*)features + (long)b * NPTS * 32;
    float fx = 0.f, fy = 0.f;
#pragma unroll
    for (int k = 0; k < KNN; ++k) {
        const int ik = __builtin_amdgcn_readlane(idx, k);   // SGPR broadcast
        const float2 v = fbase[(long)ik * 32 + lane];       // saddr + lane offset
        fx += v.x;
        fy += v.y;
    }

    // ---- 4x WMMA: D_t(16x16) = geo(16x4) @ Wtile_t(4x16) ----
    float mm[4];
#pragma unroll
    for (int t = 0; t < 4; ++t) {
        v8f c = {};
        c = __builtin_amdgcn_wmma_f32_16x16x4_f32(
                /*neg_a=*/false, a, /*neg_b=*/false, wt[t],
                /*c_mod=*/(short)0, c, /*reuse_a=*/false, /*reuse_b=*/false);
        // bias + leaky relu per element, then reduce over M (K-neighbor axis):
        // VGPR v holds M=v (lanes 0-15) / M=v+8 (lanes 16-31), same N per lane.
        float acc = 0.f;
#pragma unroll
        for (int v = 0; v < 8; ++v) {
            float x = c[v] + bv[t];
            x = (x > 0.f) ? x : (LEAKY_SLOPE * x);
            acc += x;
        }
        acc += __shfl_xor(acc, 16, 32);   // combine M=0..7 with M=8..15 halves
        mm[t] = acc * (1.0f / 16.0f);
    }

    // Store MLP means: lanes 0-15 write tiles 0,1; lanes 16-31 write tiles 2,3.
    {
        const long obase = (long)p * 128;
        const int t0 = khi * 2;
        out[obase + (t0    ) * 16 + nn] = mm[t0];
        out[obase + (t0 + 1) * 16 + nn] = mm[t0 + 1];
    }

    // Store feature means: channels 64..127 as float2 per lane.
    float2 r;
    r.x = fx * (1.0f / 16.0f);
    r.y = fy * (1.0f / 16.0f);
    ((float2*)out)[(long)p * 64 + 32 + lane] = r;
}

extern "C" void kernel_launch(void* const* d_in, const int* in_sizes, int n_in,
                              void* d_out, int out_size, void* d_ws, size_t ws_size,
                              hipStream_t stream) {
    const float* points   = (const float*)d_in[0];
    const float* features = (const float*)d_in[1];
    const int*   knn      = (const int*)  d_in[2];
    const float* W        = (const float*)d_in[3];
    const float* bias     = (const float*)d_in[4];
    float* out            = (float*)d_out;

    // 65536 points, 1 wave (32 threads) per point, 8 waves (256 threads) per block
    const int nPoints = NB * NPTS;              // 65536
    dim3 block(256);
    dim3 grid(nPoints / 8);                     // 8192 blocks
    lfa_kernel<<<grid, block, 0, stream>>>(points, features, knn, W, bias, out);
}